// SeesawGHMc_38671885533689
// MI455X (gfx1250) — compile-verified
//
#include <hip/hip_runtime.h>
#include <hip/hip_bf16.h>
#include <stdint.h>

// GHM-C loss, N=32768 rows x C=1000 classes, fp32.
//  Pass 0: zero counters.
//  Pass F: per-row target correction (move target element between bins).
//  Pass 1: pure streaming histogram, packed 6-bit per-thread counters,
//          sigmoid via TRANS-pipe v_exp/v_rcp (co-executes with VALU).
//  Pass L: log(counts).
//  Pass 2: one block/row, async-DMA row into LDS (ASYNCcnt path),
//          max-pass + sum-pass logsumexp, scalar atomic accumulate.

static constexpr int BINS = 10;
static constexpr int C    = 1000;   // reference fixes C = 1000 (divisible by 4)
static constexpr int CV   = C / 4;  // float4 chunks per row = 250

// b = min((int)(10 * sigmoid(y)), 9)  -- exactly the reference's binning.
// exp + rcp are TRANS ops: dual-issue alongside VALU on CDNA5.
__device__ __forceinline__ int ghm_bin(float y) {
  const float g = __builtin_amdgcn_rcpf(1.0f + __expf(-y));  // sigmoid(y)
  const int b = (int)(g * 10.0f);
  return b > 9 ? 9 : b;
}

__global__ void ghm_init(unsigned* __restrict__ cnt, float* __restrict__ out) {
  if (threadIdx.x < BINS) cnt[threadIdx.x] = 0u;
  if (threadIdx.x == 0) *out = 0.0f;
}

// Target correction: the main pass bins every element as g = sigmoid(x);
// the N target elements actually use g = 1 - sigmoid(x) = sigmoid(-x).
__global__ __launch_bounds__(256) void ghm_fix(const float* __restrict__ x,
                                               const int* __restrict__ tgt,
                                               unsigned* __restrict__ cnt,
                                               int nrows) {
  __shared__ int d[BINS];
  const int tid = threadIdx.x;
  if (tid < BINS) d[tid] = 0;
  __syncthreads();
  const int row = blockIdx.x * blockDim.x + tid;
  if (row < nrows) {
    const int t = tgt[row];
    const float xt = x[(size_t)row * C + t];
    const int bp = ghm_bin(-xt);   // correct bin for the target element
    const int bm = ghm_bin(xt);    // bin the streaming pass charged it to
    if (bp != bm) { atomicAdd(&d[bp], 1); atomicSub(&d[bm], 1); }
  }
  __syncthreads();
  if (tid < BINS && d[tid] != 0) atomicAdd(&cnt[tid], (unsigned)d[tid]);
}

// Pass 1: pure 131MB stream. Packed 64-bit accumulator: 10 x 6-bit fields
// (needs >=2134 blocks so each thread sees <=63 elements; we launch 4096).
__global__ __launch_bounds__(256) void ghm_hist(const float4* __restrict__ x4,
                                                unsigned* __restrict__ cnt,
                                                int nvec) {
  __shared__ unsigned h[BINS];
  const int tid = threadIdx.x;
  if (tid < BINS) h[tid] = 0u;
  __syncthreads();

  unsigned long long acc = 0ull;
  const int stride = gridDim.x * blockDim.x;
  for (int v = blockIdx.x * blockDim.x + tid; v < nvec; v += stride) {
    const float4 xv = x4[v];
    acc += 1ull << (6 * ghm_bin(xv.x));
    acc += 1ull << (6 * ghm_bin(xv.y));
    acc += 1ull << (6 * ghm_bin(xv.z));
    acc += 1ull << (6 * ghm_bin(xv.w));
  }

  const int lane = tid & 31;   // wave32
#pragma unroll
  for (int b = 0; b < BINS; ++b) {
    unsigned v = (unsigned)((acc >> (6 * b)) & 63ull);
#pragma unroll
    for (int off = 16; off; off >>= 1) v += __shfl_xor(v, off);
    if (lane == 0 && v) atomicAdd(&h[b], v);
  }
  __syncthreads();
  if (tid < BINS) atomicAdd(&cnt[tid], h[tid]);
}

__global__ void ghm_logc(const unsigned* __restrict__ cnt,
                         float* __restrict__ logc) {
  if (threadIdx.x < BINS)
    logc[threadIdx.x] = logf(fmaxf((float)cnt[threadIdx.x], 1.0f));
}

// Pass 2: one block per row. Row staged into LDS via gfx1250 async DMA;
// per-row log-weight table lw[b] = min(log c[b] - log c[b_t], 0);
// z_j = x_j + lw[bin(x_j)] (z_t = x_t exactly); logsumexp in two phases.
__global__ __launch_bounds__(256) void ghm_row(const float* __restrict__ x,
                                               const int* __restrict__ tgt,
                                               const float* __restrict__ logc,
                                               float* __restrict__ out,
                                               int nrows) {
  __shared__ __align__(16) float lrow[C];
  __shared__ float lw[BINS];
  __shared__ float red[8];
  __shared__ float s_M;
  __shared__ float s_xt;

  const int row = blockIdx.x;
  const int tid = threadIdx.x;
  const float* xr = x + (size_t)row * C;
  const int t = tgt[row];

  // Async stage: each active lane DMAs one b128 chunk (ASYNCcnt tracked).
  if (tid < CV) {
    const uint32_t ldst = (uint32_t)(uintptr_t)&lrow[tid * 4]; // low 32b = LDS offset
    const uint64_t ga   = (uint64_t)(uintptr_t)(xr + tid * 4);
    asm volatile("global_load_async_to_lds_b128 %0, %1, off"
                 :: "v"(ldst), "v"(ga) : "memory");
  }

  if (tid < BINS) {
    const float xt = xr[t];
    const int bt = ghm_bin(-xt);                     // target element's bin
    lw[tid] = fminf(logc[tid] - logc[bt], 0.0f);
    if (tid == 0) s_xt = xt;
  }

  asm volatile("s_wait_asynccnt 0" ::: "memory");
  __syncthreads();

  // Phase A: weighted logits (kept in registers) + block max.
  float m = -3.0e38f;
  float z4[4];
  if (tid < CV) {
    const float4 xv = *(const float4*)&lrow[tid * 4];
    const float xe[4] = {xv.x, xv.y, xv.z, xv.w};
    const int j0 = tid * 4;
#pragma unroll
    for (int e = 0; e < 4; ++e) {
      const float xx = xe[e];
      const float z = (j0 + e == t) ? xx : xx + lw[ghm_bin(xx)];
      z4[e] = z;
      m = fmaxf(m, z);
    }
  }
#pragma unroll
  for (int off = 16; off; off >>= 1) m = fmaxf(m, __shfl_xor(m, off));
  const int wave = tid >> 5;
  const int lane = tid & 31;
  if (lane == 0) red[wave] = m;
  __syncthreads();
  if (tid == 0) {
    float M = red[0];
#pragma unroll
    for (int w = 1; w < 8; ++w) M = fmaxf(M, red[w]);
    s_M = M;
  }
  __syncthreads();

  // Phase B: sum of exp(z - M).
  const float M = s_M;
  float s = 0.0f;
  if (tid < CV) {
#pragma unroll
    for (int e = 0; e < 4; ++e) s += __expf(z4[e] - M);
  }
#pragma unroll
  for (int off = 16; off; off >>= 1) s += __shfl_xor(s, off);
  if (lane == 0) red[wave] = s;
  __syncthreads();
  if (tid == 0) {
    float S = 0.0f;
#pragma unroll
    for (int w = 0; w < 8; ++w) S += red[w];
    const float loss_row = M + logf(S) - s_xt;   // logsumexp - z_target
    atomicAdd(out, loss_row * (1.0f / (float)nrows));
  }
}

extern "C" void kernel_launch(void* const* d_in, const int* in_sizes, int n_in,
                              void* d_out, int out_size, void* d_ws, size_t ws_size,
                              hipStream_t stream) {
  const float* x   = (const float*)d_in[0];
  const int*   tgt = (const int*)d_in[1];
  float*       out = (float*)d_out;

  const int total = in_sizes[0];          // N*C = 32768000
  const int nrows = in_sizes[1];          // N   = 32768
  const int nvec  = total >> 2;

  unsigned* cnt  = (unsigned*)d_ws;                    // 10 bin counters
  float*    logc = (float*)((char*)d_ws + 64);         // 10 log-counts

  ghm_init<<<1, 32, 0, stream>>>(cnt, out);
  ghm_fix<<<(nrows + 255) / 256, 256, 0, stream>>>(x, tgt, cnt, nrows);
  ghm_hist<<<4096, 256, 0, stream>>>((const float4*)x, cnt, nvec);
  ghm_logc<<<1, 32, 0, stream>>>(cnt, logc);
  ghm_row<<<nrows, 256, 0, stream>>>(x, tgt, logc, out, nrows);
}